// GAQN_Actor_26070451487161
// MI455X (gfx1250) — compile-verified
//
#include <hip/hip_runtime.h>
#include <hip/hip_bf16.h>
#include <stdint.h>

#define EPS_C 0.05f

// ---------- order-preserving float <-> uint mapping (for atomic max on f32) ----------
__device__ __forceinline__ unsigned fkey(float f) {
  unsigned u = __float_as_uint(f);
  return (u & 0x80000000u) ? ~u : (u | 0x80000000u);
}

// ---------- exact JAX threefry2x32, key = PRNGKey(42) = {0, 42} ----------
__device__ __forceinline__ unsigned rotl32(unsigned x, unsigned r) {
  return (x << r) | (x >> (32u - r));
}

__device__ __forceinline__ void threefry_42(unsigned x0, unsigned x1,
                                            unsigned& o0, unsigned& o1) {
  const unsigned ks0 = 0u;
  const unsigned ks1 = 42u;
  const unsigned ks2 = 0x1BD11BDAu ^ 0u ^ 42u;
  x0 += ks0; x1 += ks1;
#define TF_G(a, b, c, d)                        \
  x0 += x1; x1 = rotl32(x1, a); x1 ^= x0;       \
  x0 += x1; x1 = rotl32(x1, b); x1 ^= x0;       \
  x0 += x1; x1 = rotl32(x1, c); x1 ^= x0;       \
  x0 += x1; x1 = rotl32(x1, d); x1 ^= x0;
  TF_G(13, 15, 26, 6)   x0 += ks1; x1 += ks2 + 1u;
  TF_G(17, 29, 16, 24)  x0 += ks2; x1 += ks0 + 2u;
  TF_G(13, 15, 26, 6)   x0 += ks0; x1 += ks1 + 3u;
  TF_G(17, 29, 16, 24)  x0 += ks1; x1 += ks2 + 4u;
  TF_G(13, 15, 26, 6)   x0 += ks2; x1 += ks0 + 5u;
#undef TF_G
  o0 = x0; o1 = x1;
}

// gumbel = -log(-log(uniform)) with JAX's bit->float recipe, minval = tiny
__device__ __forceinline__ float gumbel_from_bits(unsigned bits) {
  const float tiny = 1.17549435e-38f;
  float f = __fadd_rn(__uint_as_float((bits >> 9) | 0x3f800000u), -1.0f); // [0,1)
  float u = fmaxf(tiny, __fadd_rn(__fmul_rn(f, __fadd_rn(1.0f, -tiny)), tiny));
  return -logf(-logf(u));
}

// identical arithmetic in the smax pass and the winner pass (locked with _rn ops)
__device__ __forceinline__ float score_elem(float v, unsigned vmax_key,
                                            float cntf, float g) {
  float probs = __fadd_rn((fkey(v) == vmax_key) ? 1.0f : 0.0f,
                          EPS_C / __fadd_rn(cntf, -1.0f));
  if (__builtin_isinf(probs)) probs = 1.0f;
  return __fadd_rn(logf(probs), g);
}

// ---------- wave32 segmented-run aggregation (batch_idx is sorted) ----------
// Hillis-Steele inclusive segmented scans over contiguous runs of equal `b`;
// only the run-tail lane issues the global atomic (~30x fewer L2 atomic RMWs).
// Shuffles lower to ds_bpermute_b32 (LDS crossbar, no memory).

__device__ __forceinline__ void wave_seg_cnt_max(unsigned* cnt, unsigned* vmaxk,
                                                 int b, unsigned key,
                                                 bool last_elem) {
  const int lane = threadIdx.x & 31;
  unsigned c = 1u;
  unsigned v = key;
#pragma unroll
  for (int off = 1; off < 32; off <<= 1) {
    int bo      = __shfl_up(b, off, 32);
    unsigned co = __shfl_up(c, off, 32);
    unsigned vo = __shfl_up(v, off, 32);
    if (lane >= off && bo == b) {
      c += co;
      v = (v > vo) ? v : vo;
    }
  }
  int bn = __shfl_down(b, 1, 32);
  if (lane == 31 || last_elem || bn != b) {
    atomicAdd(cnt + b, c);
    atomicMax(vmaxk + b, v);
  }
}

__device__ __forceinline__ void wave_seg_umax(unsigned* arr, int b, unsigned key,
                                              bool last_elem) {
  const int lane = threadIdx.x & 31;
  unsigned v = key;
#pragma unroll
  for (int off = 1; off < 32; off <<= 1) {
    int bo      = __shfl_up(b, off, 32);
    unsigned vo = __shfl_up(v, off, 32);
    if (lane >= off && bo == b) v = (v > vo) ? v : vo;
  }
  int bn = __shfl_down(b, 1, 32);
  if (lane == 31 || last_elem || bn != b) atomicMax(arr + b, v);
}

__device__ __forceinline__ void wave_seg_imax(int* arr, int b, int val,
                                              bool last_elem) {
  const int lane = threadIdx.x & 31;
  int v = val;
#pragma unroll
  for (int off = 1; off < 32; off <<= 1) {
    int bo = __shfl_up(b, off, 32);
    int vo = __shfl_up(v, off, 32);
    if (lane >= off && bo == b) v = (v > vo) ? v : vo;
  }
  int bn = __shfl_down(b, 1, 32);
  if ((lane == 31 || last_elem || bn != b) && v >= 0) atomicMax(arr + b, v);
}

// ---------- kernels ----------
__global__ void k_init(unsigned* cnt, unsigned* vmaxk, unsigned* smaxk,
                       int* winner, int B) {
  int b = blockIdx.x * blockDim.x + threadIdx.x;
  if (b < B) {
    cnt[b]    = 0u;
    vmaxk[b]  = 0u;   // fkey(-inf) = 0x007FFFFF > 0, so 0 is below every real value
    smaxk[b]  = 0u;
    winner[b] = -1;
  }
}

__global__ void k_cnt_vmax(const float* __restrict__ values,
                           const int* __restrict__ bidx,
                           unsigned* cnt, unsigned* vmaxk, int N) {
  int i = blockIdx.x * blockDim.x + threadIdx.x;
  if (i >= N) return;
  if (i + 2048 < N) __builtin_prefetch(&values[i + 2048], 0, 0); // global_prefetch_b8
  int b = bidx[i];
  wave_seg_cnt_max(cnt, vmaxk, b, fkey(values[i]), i == N - 1);
}

__global__ void k_score_smax(const float* __restrict__ values,
                             const int* __restrict__ bidx,
                             const unsigned* __restrict__ cnt,
                             const unsigned* __restrict__ vmaxk,
                             unsigned* smaxk, int half) {
  int p = blockIdx.x * blockDim.x + threadIdx.x;
  if (p >= half) return;
  unsigned g0, g1;
  threefry_42((unsigned)p, (unsigned)(p + half), g0, g1); // JAX splits iota at N/2
  {
    int b = bidx[p];
    float s = score_elem(values[p], vmaxk[b], (float)cnt[b], gumbel_from_bits(g0));
    wave_seg_umax(smaxk, b, fkey(s), p == half - 1);
  }
  {
    int i = p + half;
    int b = bidx[i];
    float s = score_elem(values[i], vmaxk[b], (float)cnt[b], gumbel_from_bits(g1));
    wave_seg_umax(smaxk, b, fkey(s), p == half - 1);
  }
}

__global__ void k_winner(const float* __restrict__ values,
                         const int* __restrict__ bidx,
                         const unsigned* __restrict__ cnt,
                         const unsigned* __restrict__ vmaxk,
                         const unsigned* __restrict__ smaxk,
                         int* winner, int half) {
  int p = blockIdx.x * blockDim.x + threadIdx.x;
  if (p >= half) return;
  unsigned g0, g1;
  threefry_42((unsigned)p, (unsigned)(p + half), g0, g1);
  {
    int b = bidx[p];
    float s = score_elem(values[p], vmaxk[b], (float)cnt[b], gumbel_from_bits(g0));
    wave_seg_imax(winner, b, (fkey(s) == smaxk[b]) ? p : -1, p == half - 1);
  }
  {
    int i = p + half;
    int b = bidx[i];
    float s = score_elem(values[i], vmaxk[b], (float)cnt[b], gumbel_from_bits(g1));
    wave_seg_imax(winner, b, (fkey(s) == smaxk[b]) ? i : -1, p == half - 1);
  }
}

// Final gather: out[b] = candidates[winner[B-1-b]], 256B per row.
// DMA-style copy through LDS with the gfx1250 async global<->LDS path
// (ASYNCcnt), never staging data in VGPRs. One wave32 moves two rows:
// 32 lanes x 16B = 512B per wave.
__global__ void k_gather_async(const float* __restrict__ cand,
                               const int* __restrict__ winner,
                               float* __restrict__ out, int B, int N) {
  __shared__ __align__(16) unsigned char smem[8 * 512]; // 8 waves/block
  int lane        = threadIdx.x & 31;
  int wave_in_blk = threadIdx.x >> 5;
  int wave_global = blockIdx.x * 8 + wave_in_blk;
  int r_out = wave_global * 2 + (lane >> 4);
  int c     = lane & 15; // 16B chunk within the 256B row
  if (r_out >= B) return;

  int w = winner[B - 1 - r_out];
  if (w < 0) w = 0;            // JAX OOB gather clamps; empty segment -> row 0
  if (w >= N) w = N - 1;

  unsigned lds = (unsigned)(uintptr_t)(void*)&smem[wave_in_blk * 512 + lane * 16];
  unsigned long long gsrc = (unsigned long long)(uintptr_t)cand +
                            ((unsigned long long)w << 8) + ((unsigned)c << 4);
  unsigned long long gdst = (unsigned long long)(uintptr_t)out +
                            ((unsigned long long)r_out << 8) + ((unsigned)c << 4);

  asm volatile(
      "global_load_async_to_lds_b128 %0, %1, off\n\t"
      "s_wait_asynccnt 0x0\n\t"
      "global_store_async_from_lds_b128 %2, %0, off\n\t"
      "s_wait_asynccnt 0x0"
      :
      : "v"(lds), "v"(gsrc), "v"(gdst)
      : "memory");
}

extern "C" void kernel_launch(void* const* d_in, const int* in_sizes, int n_in,
                              void* d_out, int out_size, void* d_ws, size_t ws_size,
                              hipStream_t stream) {
  (void)n_in; (void)ws_size;
  const float* cand   = (const float*)d_in[0];
  const float* values = (const float*)d_in[1];
  const int*   bidx   = (const int*)d_in[2];
  float*       out    = (float*)d_out;

  const int D = 64;
  const int N = in_sizes[1];
  const int B = out_size / D;

  unsigned char* ws = (unsigned char*)d_ws;
  unsigned* cnt    = (unsigned*)ws;
  unsigned* vmaxk  = cnt + B;
  unsigned* smaxk  = vmaxk + B;
  int*      winner = (int*)(smaxk + B);   // total 4*B*4 = 256 KB of scratch

  k_init<<<(B + 255) / 256, 256, 0, stream>>>(cnt, vmaxk, smaxk, winner, B);
  k_cnt_vmax<<<(N + 255) / 256, 256, 0, stream>>>(values, bidx, cnt, vmaxk, N);

  const int half = N / 2;
  k_score_smax<<<(half + 255) / 256, 256, 0, stream>>>(values, bidx, cnt, vmaxk,
                                                       smaxk, half);
  k_winner<<<(half + 255) / 256, 256, 0, stream>>>(values, bidx, cnt, vmaxk,
                                                   smaxk, winner, half);

  const int waves  = (B + 1) / 2;        // 2 rows per wave32
  const int blocks = (waves + 7) / 8;    // 8 waves per 256-thread block
  k_gather_async<<<blocks, 256, 0, stream>>>(cand, winner, out, B, N);
}